// CUDAOptimizedBKCore_77936476553719
// MI455X (gfx1250) — compile-verified
//
#include <hip/hip_runtime.h>
#include <hip/hip_bf16.h>
#include <stdint.h>

// B=131072 rows, N=80. HBM-bound (~126 MB moved, ~5.4 us floor at 23.3 TB/s).
// CDNA5 path: async global->LDS B128 staging (ASYNCcnt) + padded-LDS wave32 scans.

constexpr int Ncol = 80;
constexpr int ROWS = 64;       // rows per block == blockDim.x
constexpr int ASTR = 84;       // a_tile stride: 16B-aligned rows for async B128
constexpr int PSTR = 81;       // phi/num stride: conflict-free for wave32 reads

__global__ __launch_bounds__(ROWS)
void bk_continuant_kernel(const float* __restrict__ he,     // (B, N)
                          const float* __restrict__ h0diag, // (N)
                          const float* __restrict__ h0sub,  // (N-1)
                          const float* __restrict__ h0sup,  // (N-1)
                          float* __restrict__ out,          // (B, N, 2)
                          int Btot) {
    __shared__ __align__(16) float a_tile[ROWS * ASTR]; // he tile (h0_diag added on read)
    __shared__ float sr[ROWS * PSTR];                   // phi_r[k+1] then num_r[k]
    __shared__ float si[ROWS * PSTR];                   // phi_i[k+1] then num_i[k]
    __shared__ float h0d[Ncol];
    __shared__ float bcs[Ncol - 1];
    __shared__ float den_r[ROWS], den_i[ROWS], den_inv[ROWS];

    const int tid = threadIdx.x;
    const size_t rowBase = (size_t)blockIdx.x * ROWS;
    const int rows = (Btot - (int)rowBase) < ROWS ? (Btot - (int)rowBase) : ROWS;

    // ---- Async global -> LDS staging, 16B chunks (20 per row), fully coalesced ----
    {
        const uint64_t gbase = (uint64_t)(uintptr_t)(he + rowBase * Ncol);
        const int nchunk = rows * (Ncol / 4);
        for (int i = tid; i < nchunk; i += ROWS) {
            const int r  = i / (Ncol / 4);
            const int c4 = i - r * (Ncol / 4);
            const unsigned lds_addr = (unsigned)(uintptr_t)&a_tile[r * ASTR + c4 * 4];
            const unsigned goff     = (unsigned)(i * 16);
            asm volatile("global_load_async_to_lds_b128 %0, %1, %2"
                         :: "v"(lds_addr), "v"(goff), "s"(gbase)
                         : "memory");
        }
    }

    // Small per-column tables (overlap with async tile copy).
    if (tid < Ncol)     h0d[tid] = h0diag[tid];
    if (tid < Ncol - 1) bcs[tid] = h0sup[tid] * h0sub[tid];

    asm volatile("s_wait_asynccnt 0" ::: "memory");
    __syncthreads();

    // ---- Per-lane scans: one row per thread ----
    if (tid < rows) {
        const int aoff = tid * ASTR;
        const int off  = tid * PSTR;
        #define AVAL(k) (a_tile[aoff + (k)] + h0d[(k)])

        // Backward: phi[80]=1, phi[79]=a[79]-i ; phi[t]=(a[t]-i)phi[t+1]-bc[t]phi[t+2]
        // Store slot[k] = phi[k+1] for k=0..79.
        float p_r = 1.0f, p_i = 0.0f;            // phi[t+2]
        float c_r = AVAL(Ncol - 1), c_i = -1.0f; // phi[t+1]
        sr[off + Ncol - 1] = 1.0f; si[off + Ncol - 1] = 0.0f;
        sr[off + Ncol - 2] = c_r;  si[off + Ncol - 2] = c_i;
        #pragma unroll 4
        for (int t = Ncol - 2; t >= 1; --t) {
            const float at = AVAL(t);
            const float b  = bcs[t];
            const float nr = at * c_r + c_i - b * p_r;
            const float ni = at * c_i - c_r - b * p_i;
            sr[off + t - 1] = nr; si[off + t - 1] = ni;
            p_r = c_r; p_i = c_i; c_r = nr; c_i = ni;
        }

        // Forward: th[0]=1, th[1]=a[0]-i ; num_k = th[k]*phi[k+1], overwrite slot in place.
        float tp_r = 1.0f, tp_i = 0.0f;      // th[k]
        float tc_r = AVAL(0), tc_i = -1.0f;  // th[k+1]
        #pragma unroll 4
        for (int k = 0; k < Ncol; ++k) {
            const float phr = sr[off + k];
            const float phi = si[off + k];
            const float numr = tp_r * phr - tp_i * phi;
            const float numi = tp_r * phi + tp_i * phr;
            sr[off + k] = numr; si[off + k] = numi;
            if (k < Ncol - 1) {
                const float ak = AVAL(k + 1);
                const float b  = bcs[k];
                const float nr = ak * tc_r + tc_i - b * tp_r;
                const float ni = ak * tc_i - tc_r - b * tp_i;
                tp_r = tc_r; tp_i = tc_i; tc_r = nr; tc_i = ni;
            }
        }
        den_r[tid] = tc_r;
        den_i[tid] = tc_i;
        den_inv[tid] = 1.0f / (tc_r * tc_r + tc_i * tc_i);
        #undef AVAL
    }
    __syncthreads();

    // ---- Coalesced float4 writeback (2 complex per store): g = num*conj(den)/|den|^2 ----
    float4* o4 = reinterpret_cast<float4*>(out) + rowBase * (Ncol / 2);
    const int nquad = rows * (Ncol / 2);
    for (int i = tid; i < nquad; i += ROWS) {
        const int r = i / (Ncol / 2);
        const int q = i - r * (Ncol / 2);
        const int k = 2 * q;
        const float dr = den_r[r], di = den_i[r], inv = den_inv[r];
        const float nr0 = sr[r * PSTR + k],     ni0 = si[r * PSTR + k];
        const float nr1 = sr[r * PSTR + k + 1], ni1 = si[r * PSTR + k + 1];
        float4 v;
        v.x = (nr0 * dr + ni0 * di) * inv;
        v.y = (ni0 * dr - nr0 * di) * inv;
        v.z = (nr1 * dr + ni1 * di) * inv;
        v.w = (ni1 * dr - nr1 * di) * inv;
        o4[i] = v;
    }
}

extern "C" void kernel_launch(void* const* d_in, const int* in_sizes, int n_in,
                              void* d_out, int out_size, void* d_ws, size_t ws_size,
                              hipStream_t stream) {
    const float* he    = (const float*)d_in[0];  // he_diag (B, N)
    const float* h0d   = (const float*)d_in[1];  // h0_diag (N)
    const float* h0sub = (const float*)d_in[2];  // h0_sub  (N-1)
    const float* h0sup = (const float*)d_in[3];  // h0_super(N-1)
    float* out = (float*)d_out;

    const int Btot = in_sizes[0] / Ncol;
    const int grid = (Btot + ROWS - 1) / ROWS;
    bk_continuant_kernel<<<grid, ROWS, 0, stream>>>(he, h0d, h0sub, h0sup, out, Btot);
}